// Model_52510270161002
// MI455X (gfx1250) — compile-verified
//
#include <hip/hip_runtime.h>

typedef __attribute__((ext_vector_type(2))) float v2f;
typedef __attribute__((ext_vector_type(8))) float v8f;

#define B_N     4096
#define A_N     8
#define D_N     11
#define D_PAD   12
#define P_N     3584
#define ATOM    512
#define NSAMP   4096
#define M_ROWS  (B_N * A_N)      // 32768 rows of the (B*A, D) "A" matrix
#define N_TILES (P_N / 16)       // 224
#define M_TILES (M_ROWS / 16)    // 2048
#define WAVES_PER_BLOCK 4

// ---------------------------------------------------------------------------
// Stage 1: normalize positions rows (P x 11) -> padded (P x 12), last col = 0.
// ---------------------------------------------------------------------------
__global__ void __launch_bounds__(256)
pos_normalize_kernel(const float* __restrict__ pos, float* __restrict__ posPad) {
    int p = blockIdx.x * blockDim.x + threadIdx.x;
    if (p >= P_N) return;
    float v[D_N];
    float s = 0.f;
#pragma unroll
    for (int d = 0; d < D_N; ++d) {
        v[d] = pos[p * D_N + d];
        s += v[d] * v[d];
    }
    float inv = 1.0f / sqrtf(s);
#pragma unroll
    for (int d = 0; d < D_N; ++d) posPad[p * D_PAD + d] = v[d] * inv;
    posPad[p * D_PAD + D_N] = 0.f;   // K padding for the 3rd WMMA step
}

// ---------------------------------------------------------------------------
// Stage 2: sim = index(32768x11) . posN(11x3584)^T via v_wmma_f32_16x16x4_f32,
// fused with per-row argmax over the 3584 positions.
// One wave handles one 16-row M tile across all 224 N tiles.
//
// A-frag layout (32-bit 16x4): lane l -> row (l&15); half (l>>4) holds K pair
// {2h, 2h+1} in its two VGPRs. B-frag symmetric with N = (l&15).
// C/D layout: VGPR j -> row j + 8*(l>>4), column (l&15).
// Index normalization is skipped: it scales rows uniformly, argmax-invariant.
// ---------------------------------------------------------------------------
__global__ void __launch_bounds__(WAVES_PER_BLOCK * 32)
sim_argmax_kernel(const float* __restrict__ idx,
                  const float* __restrict__ posPad,
                  int* __restrict__ bestOutWs,
                  int* __restrict__ bestOutFinal) {
    const int lane  = threadIdx.x & 31;
    const int wave  = threadIdx.x >> 5;
    const int mTile = blockIdx.x * WAVES_PER_BLOCK + wave;
    const int mBase = mTile * 16;
    const int mrow  = lane & 15;      // row within tile (or column for B)
    const int khalf = lane >> 4;      // which K pair this half-wave holds
    const int row   = mBase + mrow;

    // Preload the three A fragments (K = 11 padded to 12, zero-filled).
    v2f afrag[3];
#pragma unroll
    for (int kc = 0; kc < 3; ++kc) {
        int k = kc * 4 + khalf * 2;
        float a0 = (k < D_N)     ? idx[row * D_N + k]     : 0.f;
        float a1 = (k + 1 < D_N) ? idx[row * D_N + k + 1] : 0.f;
        afrag[kc].x = a0;
        afrag[kc].y = a1;
    }

    float bestV[8];
    int   bestP[8];
#pragma unroll
    for (int j = 0; j < 8; ++j) { bestV[j] = -3.402823466e38f; bestP[j] = 0; }

    for (int nt = 0; nt < N_TILES; ++nt) {
        const int p = nt * 16 + mrow;          // this lane's column / position
        v8f acc = {};
#pragma unroll
        for (int kc = 0; kc < 3; ++kc) {
            int k = kc * 4 + khalf * 2;        // even -> 8B-aligned float2 load
            const float2 b2 = *reinterpret_cast<const float2*>(posPad + p * D_PAD + k);
            v2f bfrag;
            bfrag.x = b2.x;
            bfrag.y = b2.y;
            acc = __builtin_amdgcn_wmma_f32_16x16x4_f32(
                false, afrag[kc], false, bfrag, (short)0, acc, false, false);
        }
        // Branch-free running argmax (keeps EXEC all-ones for the WMMA loop).
        // Strict '>' with ascending p preserves first-max semantics.
#pragma unroll
        for (int j = 0; j < 8; ++j) {
            bool gt = acc[j] > bestV[j];
            bestV[j] = gt ? acc[j] : bestV[j];
            bestP[j] = gt ? p      : bestP[j];
        }
    }

    // Cross-lane reduction: xor 8,4,2,1 stays within each 16-lane half, which
    // is exactly the set of lanes sharing rows {mBase + 8*khalf + j}.
#pragma unroll
    for (int j = 0; j < 8; ++j) {
        float v  = bestV[j];
        int   pi = bestP[j];
#pragma unroll
        for (int off = 8; off >= 1; off >>= 1) {
            float ov = __shfl_xor(v, off, 32);
            int   oi = __shfl_xor(pi, off, 32);
            bool take = (ov > v) || (ov == v && oi < pi);  // first-max tiebreak
            v  = take ? ov : v;
            pi = take ? oi : pi;
        }
        bestV[j] = v;
        bestP[j] = pi;
    }

    if (mrow == 0) {
#pragma unroll
        for (int j = 0; j < 8; ++j) {
            int r = mBase + khalf * 8 + j;
            bestOutWs[r]    = bestP[j];
            bestOutFinal[r] = bestP[j];
        }
    }
}

// ---------------------------------------------------------------------------
// Stage 3: overlap-add as an atomic-free gather. One block per batch row b:
// stage all 8 atoms (16 KB) in LDS, then each output sample sums the atoms
// whose window covers it. Every output element written exactly once -> the
// 64 MB store stream is the only HBM cost (~2.7 us at 23.3 TB/s).
// ---------------------------------------------------------------------------
__global__ void __launch_bounds__(256)
overlap_add_kernel(const float* __restrict__ atoms,
                   const int* __restrict__ intIndex,
                   float* __restrict__ out) {
    __shared__ float satoms[A_N * ATOM];
    __shared__ int   soff[A_N];
    const int b   = blockIdx.x;
    const int tid = threadIdx.x;

    for (int i = tid; i < A_N * ATOM; i += 256) satoms[i] = atoms[i];
    if (tid < A_N) soff[tid] = intIndex[b * A_N + tid];
    __syncthreads();

    for (int t = tid; t < NSAMP; t += 256) {
        float sum = 0.f;
#pragma unroll
        for (int a = 0; a < A_N; ++a) {
            unsigned rel = (unsigned)(t - soff[a]);
            if (rel < ATOM) sum += satoms[a * ATOM + rel];
        }
        out[(size_t)b * NSAMP + t] = sum;
    }
}

// ---------------------------------------------------------------------------
// d_out layout (outputs concatenated in return order):
//   [0, 32768)                      int_index as int32 (B*A)
//   [32768, 32768+16777216)         output   float32  (B*NSAMP)
//   [.., +360448)                   index pass-through float32 (B*A*D)
// ---------------------------------------------------------------------------
extern "C" void kernel_launch(void* const* d_in, const int* in_sizes, int n_in,
                              void* d_out, int out_size, void* d_ws, size_t ws_size,
                              hipStream_t stream) {
    const float* index     = (const float*)d_in[0];   // (B, A, D)
    const float* positions = (const float*)d_in[1];   // (P, D)
    const float* atoms     = (const float*)d_in[2];   // (A, ATOM)

    // Workspace: normalized+padded positions, then int_index scratch.
    float* posPad = (float*)d_ws;                                   // 172 KB
    int*   wsIdx  = (int*)((char*)d_ws + (size_t)P_N * D_PAD * sizeof(float)); // 128 KB

    int*   outIdx       = (int*)d_out;
    float* outF         = (float*)d_out;
    float* outSignal    = outF + M_ROWS;
    float* outIndexCopy = outSignal + (size_t)B_N * NSAMP;

    // index pass-through (independent of everything else)
    hipMemcpyAsync(outIndexCopy, index, (size_t)M_ROWS * D_N * sizeof(float),
                   hipMemcpyDeviceToDevice, stream);

    pos_normalize_kernel<<<(P_N + 255) / 256, 256, 0, stream>>>(positions, posPad);

    sim_argmax_kernel<<<M_TILES / WAVES_PER_BLOCK, WAVES_PER_BLOCK * 32, 0, stream>>>(
        index, posPad, wsIdx, outIdx);

    overlap_add_kernel<<<B_N, 256, 0, stream>>>(atoms, wsIdx, outSignal);
}